// Dup_58377195487815
// MI455X (gfx1250) — compile-verified
//
#include <hip/hip_runtime.h>

typedef __attribute__((ext_vector_type(2))) float v2f;
typedef __attribute__((ext_vector_type(4))) float v4f;
typedef __attribute__((ext_vector_type(8))) float v8f;

#define Ndim 4
#define Cdim 128
#define Hdim 128
#define Wdim 128
#define Kdim 1152   /* C * 3 * 3 */
#define NCH  8      /* 2 * KH * KW offset channels */

__device__ __forceinline__ int iclamp(int v, int lo, int hi) {
  return v < lo ? lo : (v > hi ? hi : v);
}

// ---------------------------------------------------------------------------
// Kernel 1: 3x3 SAME conv producing the 8 offset channels, via fp32 WMMA.
// GEMM view: D[ch][pix] (16x16, 8 chans used) = sum_K A[ch][K] * B[K][pix],
// K = c*9 + kh*3 + kw, accumulated in 288 chained V_WMMA_F32_16X16X4_F32.
// One workgroup = one (n, y) row; 8 waves each own a 16-pixel strip.
// x tile LDS layout is [kh-row][cc][col] so all staging indices are shifts.
// ---------------------------------------------------------------------------
__global__ __launch_bounds__(256) void deform_off_conv(
    const float* __restrict__ x, const float* __restrict__ w,
    const float* __restrict__ b, float* __restrict__ off) {
  __shared__ float w_lds[NCH * Kdim];      // 36 KB: full weight tensor
  __shared__ float x_lds[3 * 16 * Wdim];   // 24 KB: 3 rows x 16 ch x 128 cols

  const int tid  = threadIdx.x;
  const int wave = tid >> 5;
  const int lane = tid & 31;
  const int n    = blockIdx.x / Hdim;
  const int y    = blockIdx.x % Hdim;
  const int x0   = wave * 16;              // pixel strip base (column)
  const int pix  = lane & 15;              // M index (both lane halves)
  const int kofs = (lane < 16) ? 0 : 2;    // ISA 16x4 f32 A/B lane->K split

  // Stage all weights once, as float4 (2304 = 9*256 vectors, exact).
  {
    const v4f* ws = (const v4f*)w;
    v4f* wd = (v4f*)w_lds;
#pragma unroll
    for (int j = 0; j < 9; ++j) wd[tid + 256 * j] = ws[tid + 256 * j];
  }

  v8f acc = {};
  const float* xn = x + (size_t)n * Cdim * Hdim * Wdim;
  const int col4 = (tid & 31) * 4;         // 32 float4 per 128-col row

  for (int cb = 0; cb < Cdim / 16; ++cb) {
    __syncthreads();
    // Stage 16-channel x chunk: rows y-1..y+1, zero-padded at image edges.
    // 48 (r,cc) rows of 128 cols; 8 rows per pass, B128 loads/stores.
#pragma unroll
    for (int j = 0; j < 6; ++j) {
      int rowidx = (tid >> 5) + 8 * j;     // 0..47
      int r  = rowidx >> 4;                // 0..2  (kh row)
      int cc = rowidx & 15;                // local channel
      int gy = y + r - 1;
      v4f v = {};
      if (gy >= 0 && gy < Hdim)
        v = *(const v4f*)(xn + ((size_t)(cb * 16 + cc) * Hdim + gy) * Wdim + col4);
      *(v4f*)(x_lds + rowidx * Wdim + col4) = v;
    }
    __syncthreads();

    const int kbase = cb * 144;            // 16 channels * 9 taps
#pragma unroll
    for (int kk = 0; kk < 36; ++kk) {      // K-step of 4 per WMMA
      const int kl0 = kk * 4 + kofs;       // even -> 8B-aligned pair
      const int kl1 = kl0 + 1;

      // A fragment: weights, M = output channel (= pix), K = kbase+kl{0,1}.
      // Consecutive + 8B aligned -> single ds_load_b64.
      v2f a = {};
      if (pix < NCH)
        a = *(const v2f*)(w_lds + pix * Kdim + kbase + kl0);

      // B fragment: im2col patch, N = pixel (= pix), K rows kl0, kl1.
      // Decode is compile-time per kk (kofs selects between two constants).
      int cc0 = kl0 / 9, r0 = kl0 - cc0 * 9, kh0 = r0 / 3, kw0 = r0 - kh0 * 3;
      int cc1 = kl1 / 9, r1 = kl1 - cc1 * 9, kh1 = r1 / 3, kw1 = r1 - kh1 * 3;
      int col0 = x0 + pix + kw0 - 1;
      int col1 = x0 + pix + kw1 - 1;
      v2f bb;
      bb.x = (col0 >= 0 && col0 < Wdim)
                 ? x_lds[(kh0 * 16 + cc0) * Wdim + col0] : 0.0f;
      bb.y = (col1 >= 0 && col1 < Wdim)
                 ? x_lds[(kh1 * 16 + cc1) * Wdim + col1] : 0.0f;

      acc = __builtin_amdgcn_wmma_f32_16x16x4_f32(
          /*neg_a=*/false, a, /*neg_b=*/false, bb,
          /*c_mod=*/(short)0, acc, /*reuse_a=*/false, /*reuse_b=*/false);
    }
  }

  // D layout: VGPR j, lanes 0..15 -> M=j (channel j), N=lane (pixel).
  if (lane < 16) {
    const int gx = x0 + lane;
#pragma unroll
    for (int j = 0; j < NCH; ++j) {
      off[(((size_t)n * NCH + j) * Hdim + y) * Wdim + gx] = acc[j] + b[j];
    }
  }
}

// ---------------------------------------------------------------------------
// Kernel 2: bilinear sampling at (y+dy, x+dx) for 4 kernel positions,
// pixel-shuffled into out(N, C, 2H, 2W). Coalesced gathers (lane = column),
// nontemporal v2f stores (write-once 128 MiB; keep x resident in L2).
// ---------------------------------------------------------------------------
__global__ __launch_bounds__(128) void deform_sample(
    const float* __restrict__ x, const float* __restrict__ off,
    float* __restrict__ out) {
  const int xcol = threadIdx.x;            // 0..127
  const int n    = blockIdx.x / Hdim;
  const int y    = blockIdx.x % Hdim;
  const int c0   = blockIdx.y * 32;        // channel group
  const size_t HW = (size_t)Hdim * Wdim;

  const float* xn   = x + (size_t)n * Cdim * HW;
  const float* offn = off + (size_t)n * NCH * HW + (size_t)y * Wdim + xcol;

  int   idx[4][4];
  float wt[4][4];
#pragma unroll
  for (int p = 0; p < 4; ++p) {
    float dy = offn[(size_t)(2 * p + 0) * HW];
    float dx = offn[(size_t)(2 * p + 1) * HW];
    float sy = (float)y + dy;
    float sx = (float)xcol + dx;
    float fy = floorf(sy), fx = floorf(sx);
    float wy1 = sy - fy, wx1 = sx - fx;
    float wy0 = 1.0f - wy1, wx0 = 1.0f - wx1;
    int iy0 = (int)fy, ix0 = (int)fx;
    int iy1 = iy0 + 1, ix1 = ix0 + 1;
    float vy0 = (iy0 >= 0 && iy0 < Hdim) ? 1.0f : 0.0f;
    float vy1 = (iy1 >= 0 && iy1 < Hdim) ? 1.0f : 0.0f;
    float vx0 = (ix0 >= 0 && ix0 < Wdim) ? 1.0f : 0.0f;
    float vx1 = (ix1 >= 0 && ix1 < Wdim) ? 1.0f : 0.0f;
    int cy0 = iclamp(iy0, 0, Hdim - 1), cy1 = iclamp(iy1, 0, Hdim - 1);
    int cx0 = iclamp(ix0, 0, Wdim - 1), cx1 = iclamp(ix1, 0, Wdim - 1);
    idx[p][0] = cy0 * Wdim + cx0;  wt[p][0] = wy0 * wx0 * vy0 * vx0;
    idx[p][1] = cy0 * Wdim + cx1;  wt[p][1] = wy0 * wx1 * vy0 * vx1;
    idx[p][2] = cy1 * Wdim + cx0;  wt[p][2] = wy1 * wx0 * vy1 * vx0;
    idx[p][3] = cy1 * Wdim + cx1;  wt[p][3] = wy1 * wx1 * vy1 * vx1;
  }

  float* outn = out + (size_t)n * Cdim * 4 * HW;
  for (int ci = 0; ci < 32; ++ci) {
    const int c = c0 + ci;
    const float* xp = xn + (size_t)c * HW;
    float v[4];
#pragma unroll
    for (int p = 0; p < 4; ++p) {
      v[p] = wt[p][0] * xp[idx[p][0]] + wt[p][1] * xp[idx[p][1]] +
             wt[p][2] * xp[idx[p][2]] + wt[p][3] * xp[idx[p][3]];
    }
    // p = kh*2 + kw -> out[n][c][2y+kh][2x+kw]; KW pair is contiguous.
    size_t o0 = ((size_t)c * 2 * Hdim + 2 * y) * (2 * Wdim) + 2 * xcol;
    v2f r0; r0.x = v[0]; r0.y = v[1];
    v2f r1; r1.x = v[2]; r1.y = v[3];
    __builtin_nontemporal_store(r0, (v2f*)(outn + o0));
    __builtin_nontemporal_store(r1, (v2f*)(outn + o0 + 2 * Wdim));
  }
}

extern "C" void kernel_launch(void* const* d_in, const int* in_sizes, int n_in,
                              void* d_out, int out_size, void* d_ws, size_t ws_size,
                              hipStream_t stream) {
  const float* x  = (const float*)d_in[0];
  const float* w  = (const float*)d_in[1];
  const float* b  = (const float*)d_in[2];
  float* off = (float*)d_ws;               // N*8*H*W fp32 = 2 MiB scratch
  float* out = (float*)d_out;

  deform_off_conv<<<dim3(Ndim * Hdim), dim3(256), 0, stream>>>(x, w, b, off);
  deform_sample<<<dim3(Ndim * Hdim, Cdim / 32), dim3(Wdim), 0, stream>>>(x, off, out);
}